// FokkerPlanck2DBaseConditioned_66048007078448
// MI455X (gfx1250) — compile-verified
//
#include <hip/hip_runtime.h>
#include <cstdint>

// ---------------------------------------------------------------------------
// Fokker-Planck 2D explicit update, B=32, 512x512, fp32.
// Memory-bound: 224 MiB traffic vs ~0.5 GFLOP -> ~10us floor @ 23.3 TB/s.
// CDNA5 path: stage halo tiles into LDS via the Tensor Data Mover
// (tensor_load_to_lds + s_wait_tensorcnt), compute stencils from LDS.
// Interior tiles take a block-uniform branch-free fast path with
// consecutive-x columns per thread so LDS loads CSE across the unroll.
// ---------------------------------------------------------------------------

#define NXD 512
#define NYD 512
#define TS  32
#define HALO_W (TS + 2)               // 34
#define PLANE (HALO_W * HALO_W)       // 1156 floats per staged plane
#define NPLANES 6                     // f, A0, A1, B0, B1, B2
#define THREADS 256

// ---- feature detection (device pass only) ---------------------------------
#if defined(__HIP_DEVICE_COMPILE__) && defined(__has_builtin)
#  if __has_builtin(__builtin_amdgcn_tensor_load_to_lds)
#    define HAVE_TDM 1
#  endif
#  if __has_builtin(__builtin_amdgcn_s_wait_tensorcnt)
#    define HAVE_TCNT 1
#  endif
#endif
#ifndef HAVE_TDM
#  define HAVE_TDM 0
#endif
#ifndef HAVE_TCNT
#  define HAVE_TCNT 0
#endif

#if defined(__clang_major__) && (__clang_major__ >= 23)
#  define TDM_6ARG 1   // amdgpu-toolchain / therock-10.0 headers: 6-arg builtin
#else
#  define TDM_6ARG 0   // ROCm 7.2 clang-22: 5-arg builtin
#endif

typedef __attribute__((ext_vector_type(4))) unsigned int uint4v;
typedef __attribute__((ext_vector_type(8))) int          int8v;
typedef __attribute__((ext_vector_type(4))) int          int4v;

#if HAVE_TDM
// Issue one TDM 2D tile load: global (row stride NYD elements) -> packed LDS.
// wE = tile width in elements (contiguous y), hR = tile rows (x).
__device__ __forceinline__ void tdm_load_2d(const float* gsrc, unsigned lds_off,
                                            int wE, int hR)
{
  unsigned long long ga = (unsigned long long)(uintptr_t)gsrc;

  // ---- D# group 0 (128b): count=1 | lds_addr | global_addr[56:0] | type=2
  uint4v g0;
  g0.x = 1u;                                         // count=1 (valid), user mode
  g0.y = lds_off;                                    // LDS byte address
  g0.z = (unsigned)ga;                               // global_addr[31:0]
  g0.w = (unsigned)((ga >> 32) & 0x01ffffffULL)      // global_addr[56:32]
         | 0x80000000u;                              // type=2 ("image") @127:126

  // ---- D# group 1 (256b)
  int8v g1;
  g1[0] = 0x00020000;                 // workgroup_mask=0, data_size=2 (4B), no pad
  g1[1] = (int)(NYD << 16);           // atomic_barrier_addr=0 | tensor_dim0[15:0]
  g1[2] = (int)(NXD << 16);           // tensor_dim0[31:16]=0 | tensor_dim1[15:0]
  g1[3] = (int)(wE  << 16);           // tensor_dim1[31:16]=0 | tile_dim0
  g1[4] = (int)(hR & 0xFFFF);         // tile_dim1 | tile_dim2=0 (2D)
  g1[5] = NYD;                        // tensor_dim0_stride[31:0]
  g1[6] = 0;                          // stride hi | tensor_dim1_stride lo (unused 2D)
  g1[7] = 0;

  int4v z4 = {0, 0, 0, 0};            // groups 2/3: zeros (2D tensor)
#if TDM_6ARG
  int8v z8 = {0, 0, 0, 0, 0, 0, 0, 0};
  __builtin_amdgcn_tensor_load_to_lds(g0, g1, z4, z4, z8, 0);
#else
  __builtin_amdgcn_tensor_load_to_lds(g0, g1, z4, z4, 0);
#endif
}
#endif // HAVE_TDM

// ---- general (edge-tile) LDS accessors ------------------------------------
struct Tile {
  const float* s;   // LDS base
  int x0, y0, w;    // global origin of staged region, dynamic row pitch
};

__device__ __forceinline__ float ld(const Tile& t, int p, int x, int y) {
  return t.s[p * PLANE + (x - t.x0) * t.w + (y - t.y0)];
}
// product plane value: plane p times f (plane 0)
__device__ __forceinline__ float pr(const Tile& t, int p, int x, int y) {
  return ld(t, p, x, y) * ld(t, 0, x, y);
}

// d/dx with edge_order=2 one-sided boundary stencils
__device__ __forceinline__ float gx1(const Tile& t, int p, int x, int y) {
  if (x == 0)
    return -1.5f * pr(t,p,0,y) + 2.0f * pr(t,p,1,y) - 0.5f * pr(t,p,2,y);
  if (x == NXD - 1)
    return  1.5f * pr(t,p,x,y) - 2.0f * pr(t,p,x-1,y) + 0.5f * pr(t,p,x-2,y);
  return 0.5f * (pr(t,p,x+1,y) - pr(t,p,x-1,y));
}
__device__ __forceinline__ float gy1(const Tile& t, int p, int x, int y) {
  if (y == 0)
    return -1.5f * pr(t,p,x,0) + 2.0f * pr(t,p,x,1) - 0.5f * pr(t,p,x,2);
  if (y == NYD - 1)
    return  1.5f * pr(t,p,x,y) - 2.0f * pr(t,p,x,y-1) + 0.5f * pr(t,p,x,y-2);
  return 0.5f * (pr(t,p,x,y+1) - pr(t,p,x,y-1));
}
// d2/dx2 with one-sided 2nd-order boundary stencils
__device__ __forceinline__ float gx2(const Tile& t, int p, int x, int y) {
  if (x == 0)
    return 2.0f*pr(t,p,0,y) - 5.0f*pr(t,p,1,y) + 4.0f*pr(t,p,2,y) - pr(t,p,3,y);
  if (x == NXD - 1)
    return 2.0f*pr(t,p,x,y) - 5.0f*pr(t,p,x-1,y) + 4.0f*pr(t,p,x-2,y) - pr(t,p,x-3,y);
  return pr(t,p,x+1,y) - 2.0f*pr(t,p,x,y) + pr(t,p,x-1,y);
}
__device__ __forceinline__ float gy2(const Tile& t, int p, int x, int y) {
  if (y == 0)
    return 2.0f*pr(t,p,x,0) - 5.0f*pr(t,p,x,1) + 4.0f*pr(t,p,x,2) - pr(t,p,x,3);
  if (y == NYD - 1)
    return 2.0f*pr(t,p,x,y) - 5.0f*pr(t,p,x,y-1) + 4.0f*pr(t,p,x,y-2) - pr(t,p,x,y-3);
  return pr(t,p,x,y+1) - 2.0f*pr(t,p,x,y) + pr(t,p,x,y-1);
}
// mixed: d/dx of (d/dy B2f) and d/dy of (d/dx B2f) -- boundary orders differ
__device__ __forceinline__ float gx_of_gy(const Tile& t, int x, int y) {
  if (x == 0)
    return -1.5f*gy1(t,5,0,y) + 2.0f*gy1(t,5,1,y) - 0.5f*gy1(t,5,2,y);
  if (x == NXD - 1)
    return  1.5f*gy1(t,5,x,y) - 2.0f*gy1(t,5,x-1,y) + 0.5f*gy1(t,5,x-2,y);
  return 0.5f * (gy1(t,5,x+1,y) - gy1(t,5,x-1,y));
}
__device__ __forceinline__ float gy_of_gx(const Tile& t, int x, int y) {
  if (y == 0)
    return -1.5f*gx1(t,5,x,0) + 2.0f*gx1(t,5,x,1) - 0.5f*gx1(t,5,x,2);
  if (y == NYD - 1)
    return  1.5f*gx1(t,5,x,y) - 2.0f*gx1(t,5,x,y-1) + 0.5f*gx1(t,5,x,y-2);
  return 0.5f * (gx1(t,5,x+1,y) - gx1(t,5,x-1,y));
}

// ---------------------------------------------------------------------------
__global__ __launch_bounds__(THREADS)
void fp2d_kernel(const float* __restrict__ f, const float* __restrict__ dt,
                 const float* __restrict__ A, const float* __restrict__ B,
                 float* __restrict__ out)
{
#if defined(__HIP_DEVICE_COMPILE__)
  __shared__ float smem[NPLANES * PLANE];   // 27,744 bytes

  const int b   = blockIdx.z;
  const int gx0 = blockIdx.y * TS;          // output tile origin (x)
  const int gy0 = blockIdx.x * TS;          // output tile origin (y, contiguous)

  const bool interior = (gx0 > 0) && (gx0 + TS < NXD) &&
                        (gy0 > 0) && (gy0 + TS < NYD);

  // staged region = output tile + 1-halo, clamped to domain
  const int x0 = (gx0 > 0) ? gx0 - 1 : 0;
  const int y0 = (gy0 > 0) ? gy0 - 1 : 0;
  const int x1 = (gx0 + TS < NXD) ? gx0 + TS : NXD - 1;
  const int y1 = (gy0 + TS < NYD) ? gy0 + TS : NYD - 1;
  const int w  = y1 - y0 + 1;               // row pitch in LDS (<= 34)
  const int h  = x1 - x0 + 1;               // rows (<= 34)

  const size_t plane = (size_t)NXD * NYD;
  const size_t toff  = (size_t)x0 * NYD + y0;
  const float* gp[NPLANES];
  gp[0] = f + (size_t)b * plane + toff;
  gp[1] = A + (size_t)b * 2 * plane + toff;
  gp[2] = gp[1] + plane;
  gp[3] = B + (size_t)b * 3 * plane + toff;
  gp[4] = gp[3] + plane;
  gp[5] = gp[4] + plane;

#if HAVE_TDM
  // One wave issues the 6 DMA tile loads; TDM ignores EXEC, so keep the
  // issue inside a wave-0-only branch (other waves s_cbranch_execz over it).
  if (threadIdx.x < 32) {
    const unsigned lb = (unsigned)(uintptr_t)&smem[0];   // low 32b = LDS addr
    #pragma unroll
    for (int p = 0; p < NPLANES; ++p)
      tdm_load_2d(gp[p], lb + (unsigned)(p * PLANE * 4), w, h);
#if HAVE_TCNT
    __builtin_amdgcn_s_wait_tensorcnt(0);
#else
    asm volatile("s_wait_tensorcnt 0x0" ::: "memory");
#endif
  }
#else
  // Fallback: cooperative vector-load staging.
  {
    const int n = w * h;
    for (int p = 0; p < NPLANES; ++p)
      for (int i = threadIdx.x; i < n; i += THREADS) {
        const int xx = i / w;
        const int yy = i - xx * w;
        smem[p * PLANE + i] = gp[p][(size_t)xx * NYD + yy];
      }
  }
#endif
  __syncthreads();

  const float dtv = dt[b];
  const int   ty  = threadIdx.x & (TS - 1);     // y within tile (lane-contiguous)
  const int   txb = (threadIdx.x >> 5) * 4;     // base x row: 4 consecutive rows
  float* const op = out + (size_t)b * plane;

  if (interior) {
    // ---- block-uniform fast path: pure central stencils, pitch = 34 -------
    // Local coords: lx = x - (gx0-1), ly = y - (gy0-1); outputs at lx,ly >= 1.
    const float* s = smem;
    const int ly = ty + 1;
    #pragma unroll
    for (int k = 0; k < 4; ++k) {
      const int lx = txb + 1 + k;
      const int c  = lx * HALO_W + ly;                     // center index
      #define PF(dx, dy)  s[(c) + (dx) * HALO_W + (dy)]                  // f
      #define PP(p,dx,dy) (s[(p)*PLANE + (c) + (dx)*HALO_W + (dy)] * PF(dx,dy))

      const float fC  = PF(0, 0);
      const float adv = 0.5f * (PP(1, 1, 0) - PP(1, -1, 0))            // d/dx A0f
                      + 0.5f * (PP(2, 0, 1) - PP(2, 0, -1));           // d/dy A1f
      const float dxx = PP(3, 1, 0) - 2.0f * PP(3, 0, 0) + PP(3, -1, 0);
      const float dyy = PP(4, 0, 1) - 2.0f * PP(4, 0, 0) + PP(4, 0, -1);
      // gxgy + gygx (they coincide at interior): 2 * 0.25 * cross
      const float mix = 0.5f * (PP(5, 1, 1) - PP(5, 1, -1)
                               - PP(5, -1, 1) + PP(5, -1, -1));
      #undef PP
      #undef PF

      float v = fC + dtv * (-adv + 0.5f * (dxx + dyy + mix));
      v = (v > 0.0f) ? v : 0.0f;
      op[(size_t)(gx0 + txb + k) * NYD + (gy0 + ty)] = v;
    }
  } else {
    // ---- general path for domain-edge tiles (one-sided boundary stencils) -
    const Tile t{smem, x0, y0, w};
    #pragma unroll
    for (int k = 0; k < 4; ++k) {
      const int x = gx0 + txb + k;
      const int y = gy0 + ty;

      const float adv = gx1(t, 1, x, y) + gy1(t, 2, x, y);           // div(A f)
      const float dif = gx2(t, 3, x, y) + gy2(t, 4, x, y)
                      + gx_of_gy(t, x, y) + gy_of_gx(t, x, y);       // div^2(B f)

      float v = ld(t, 0, x, y) + dtv * (-adv + 0.5f * dif);
      v = (v > 0.0f) ? v : 0.0f;                                     // clamp
      op[(size_t)x * NYD + y] = v;
    }
  }
#endif // __HIP_DEVICE_COMPILE__
}

// ---------------------------------------------------------------------------
extern "C" void kernel_launch(void* const* d_in, const int* in_sizes, int n_in,
                              void* d_out, int out_size, void* d_ws, size_t ws_size,
                              hipStream_t stream)
{
  (void)n_in; (void)out_size; (void)d_ws; (void)ws_size;

  const float* f  = (const float*)d_in[0];
  const float* dt = (const float*)d_in[1];
  const float* A  = (const float*)d_in[2];
  const float* B  = (const float*)d_in[3];
  float* out = (float*)d_out;

  const int batches = in_sizes[1];          // dt has one entry per batch
  dim3 grid(NYD / TS, NXD / TS, batches);   // 16 x 16 x 32 blocks
  hipLaunchKernelGGL(fp2d_kernel, grid, dim3(THREADS), 0, stream,
                     f, dt, A, B, out);
}